// GCN_70188355551853
// MI455X (gfx1250) — compile-verified
//
#include <hip/hip_runtime.h>
#include <hip/hip_bf16.h>

typedef float v2f __attribute__((ext_vector_type(2)));
typedef float v8f __attribute__((ext_vector_type(8)));

// ---------------------------------------------------------------------------
// Degree / normalization precompute
// ---------------------------------------------------------------------------
__global__ __launch_bounds__(256) void gcn_fill_deg(float* __restrict__ deg, int n) {
    int i = blockIdx.x * 256 + threadIdx.x;
    if (i < n) deg[i] = 1.0f;  // self-loop contributes 1 to in-degree
}

__global__ __launch_bounds__(256) void gcn_deg_count(const int* __restrict__ dst,
                                                     float* __restrict__ deg, int ne) {
    int e = blockIdx.x * 256 + threadIdx.x;
    if (e < ne) unsafeAtomicAdd(&deg[dst[e]], 1.0f);
}

__global__ __launch_bounds__(256) void gcn_dinv(float* __restrict__ deg, int n) {
    int i = blockIdx.x * 256 + threadIdx.x;
    if (i < n) {
        float d = deg[i];
        deg[i] = (d > 0.0f) ? rsqrtf(d) : 0.0f;
    }
}

__global__ __launch_bounds__(256) void gcn_coef(const int* __restrict__ src,
                                                const int* __restrict__ dst,
                                                const float* __restrict__ dinv,
                                                float* __restrict__ coef, int ne) {
    int e = blockIdx.x * 256 + threadIdx.x;
    if (e < ne) coef[e] = dinv[src[e]] * dinv[dst[e]];
}

// ---------------------------------------------------------------------------
// GEMM: out[nrows x NOUT] = in[nrows x 128] @ W[128 x NOUT]   (fp32 WMMA)
// Block: 256 threads (8 waves), 128-row tile. Wave w owns rows w*16..w*16+15
// and all NOUT/16 column tiles, accumulating V_WMMA_F32_16X16X4_F32 over K.
// ---------------------------------------------------------------------------
template <int NOUT>
__global__ __launch_bounds__(256) void gcn_gemm_wmma(const float* __restrict__ in,
                                                     const float* __restrict__ W,
                                                     float* __restrict__ out, int nrows) {
    constexpr int K   = 128;
    constexpr int AST = K + 4;   // padded LDS strides -> conflict-free fragment reads
    constexpr int WST = K + 4;
    constexpr int NT  = NOUT / 16;

    __shared__ float lA[128 * AST];     // A tile, row-major [128][K]
    __shared__ float lWt[NOUT * WST];   // W transposed, [NOUT][K]

    const int tid     = threadIdx.x;
    const int rowBase = blockIdx.x * 128;

    // Stage W transposed (coalesced global read, strided LDS write)
    for (int i = tid; i < K * NOUT; i += 256) {
        int r = i / NOUT, c = i % NOUT;          // W is row-major [K][NOUT]
        lWt[c * WST + r] = W[i];
    }
    // Stage A tile as float4 (zero-pad rows past nrows)
    for (int i = tid; i < 128 * (K / 4); i += 256) {
        int r = i / (K / 4), c4 = (i % (K / 4)) * 4;
        int grow = rowBase + r;
        float4 v = make_float4(0.f, 0.f, 0.f, 0.f);
        if (grow < nrows) v = *(const float4*)(in + (size_t)grow * K + c4);
        *(float4*)(lA + r * AST + c4) = v;
    }
    __syncthreads();

    const int wave = tid >> 5;
    const int lane = tid & 31;
    const int hw   = lane >> 4;   // half-wave select
    const int l16  = lane & 15;

    v8f acc[NT] = {};

    // A fragment (16x4 f32): lane l holds A[l%16][k + (l/16)*2 + {0,1}]
    const float* aRow  = lA + (wave * 16 + l16) * AST + hw * 2;
    // B fragment (4x16 f32): lane l holds B[k + (l/16)*2 + {0,1}][l%16]
    //   = Wt[col][k + ...] -> contiguous pair in transposed LDS layout
    const float* bBase = lWt + l16 * WST + hw * 2;

#pragma unroll 4
    for (int k = 0; k < K; k += 4) {
        v2f a = *(const v2f*)(aRow + k);
#pragma unroll
        for (int t = 0; t < NT; ++t) {
            v2f b = *(const v2f*)(bBase + t * 16 * WST + k);
            acc[t] = __builtin_amdgcn_wmma_f32_16x16x4_f32(
                false, a, false, b, (short)0, acc[t], false, false);
        }
    }

    // C/D layout: VGPR j in lane l = D[row j + (l/16)*8][col l%16]
#pragma unroll
    for (int t = 0; t < NT; ++t) {
#pragma unroll
        for (int j = 0; j < 8; ++j) {
            int grow = rowBase + wave * 16 + hw * 8 + j;
            if (grow < nrows)
                out[(size_t)grow * NOUT + t * 16 + l16] = acc[t][j];
        }
    }
}

// ---------------------------------------------------------------------------
// agg[i] = bias + h[i] * dinv[i]^2   (self-loop term; also initializes output)
// ---------------------------------------------------------------------------
template <int F>
__global__ __launch_bounds__(256) void gcn_init_agg(const float* __restrict__ h,
                                                    const float* __restrict__ bias,
                                                    const float* __restrict__ dinv,
                                                    float* __restrict__ agg, int nrows) {
    int idx   = blockIdx.x * 256 + threadIdx.x;
    int total = nrows * (F / 4);
    if (idx >= total) return;
    int row = idx / (F / 4);
    int c4  = (idx % (F / 4)) * 4;
    float di = dinv[row];
    float cf = di * di;
    float4 hv = *(const float4*)(h + (size_t)row * F + c4);
    float4 bv = *(const float4*)(bias + c4);
    float4 o;
    o.x = bv.x + hv.x * cf;
    o.y = bv.y + hv.y * cf;
    o.z = bv.z + hv.z * cf;
    o.w = bv.w + hv.w * cf;
    *(float4*)(agg + (size_t)row * F + c4) = o;
}

// ---------------------------------------------------------------------------
// Edge scatter: one wave32 per edge. Coalesced 512B (or 256B) row gather from
// h[src], fp32 atomic adds into agg[dst] (resolve in the 192MB L2).
// ---------------------------------------------------------------------------
template <int F>
__global__ __launch_bounds__(256) void gcn_scatter(const int* __restrict__ src,
                                                   const int* __restrict__ dst,
                                                   const float* __restrict__ coef,
                                                   const float* __restrict__ h,
                                                   float* __restrict__ agg, int ne) {
    constexpr int VEC = F / 32;
    int e    = blockIdx.x * 8 + (threadIdx.x >> 5);
    int lane = threadIdx.x & 31;
    if (e >= ne) return;
    int   s = src[e];
    int   d = dst[e];
    float c = coef[e];
    const float* hp = h + (size_t)s * F + lane * VEC;
    float*       ap = agg + (size_t)d * F + lane * VEC;
    if constexpr (VEC == 4) {
        float4 v = *(const float4*)hp;
        unsafeAtomicAdd(ap + 0, v.x * c);
        unsafeAtomicAdd(ap + 1, v.y * c);
        unsafeAtomicAdd(ap + 2, v.z * c);
        unsafeAtomicAdd(ap + 3, v.w * c);
    } else {
        float2 v = *(const float2*)hp;
        unsafeAtomicAdd(ap + 0, v.x * c);
        unsafeAtomicAdd(ap + 1, v.y * c);
    }
}

// ---------------------------------------------------------------------------
// Fused LayerNorm + ReLU over 128 features; one wave32 per row (4 feats/lane),
// butterfly shuffle reductions.
// ---------------------------------------------------------------------------
__global__ __launch_bounds__(256) void gcn_norm_relu(const float* __restrict__ in,
                                                     const float* __restrict__ g,
                                                     const float* __restrict__ b,
                                                     float* __restrict__ out, int nrows) {
    int row  = blockIdx.x * 8 + (threadIdx.x >> 5);
    int lane = threadIdx.x & 31;
    if (row >= nrows) return;
    float4 v = *(const float4*)(in + (size_t)row * 128 + lane * 4);

    float s = v.x + v.y + v.z + v.w;
#pragma unroll
    for (int off = 16; off; off >>= 1) s += __shfl_xor(s, off, 32);
    float mu = s * (1.0f / 128.0f);

    float dx = v.x - mu, dy = v.y - mu, dz = v.z - mu, dw = v.w - mu;
    float q = dx * dx + dy * dy + dz * dz + dw * dw;
#pragma unroll
    for (int off = 16; off; off >>= 1) q += __shfl_xor(q, off, 32);
    float r = rsqrtf(q * (1.0f / 128.0f) + 1e-5f);

    float4 gv = *(const float4*)(g + lane * 4);
    float4 bv = *(const float4*)(b + lane * 4);
    float4 o;
    o.x = fmaxf(dx * r * gv.x + bv.x, 0.0f);
    o.y = fmaxf(dy * r * gv.y + bv.y, 0.0f);
    o.z = fmaxf(dz * r * gv.z + bv.z, 0.0f);
    o.w = fmaxf(dw * r * gv.w + bv.w, 0.0f);
    *(float4*)(out + (size_t)row * 128 + lane * 4) = o;
}

// ---------------------------------------------------------------------------
// Orchestration
// ---------------------------------------------------------------------------
extern "C" void kernel_launch(void* const* d_in, const int* in_sizes, int n_in,
                              void* d_out, int out_size, void* d_ws, size_t ws_size,
                              hipStream_t stream) {
    const float* x   = (const float*)d_in[0];
    const int*   ei  = (const int*)d_in[1];   // [2, E]
    const float* W0  = (const float*)d_in[2];
    const float* b0  = (const float*)d_in[3];
    const float* W1  = (const float*)d_in[4];
    const float* b1  = (const float*)d_in[5];
    const float* W2  = (const float*)d_in[6];
    const float* b2  = (const float*)d_in[7];
    const float* g0  = (const float*)d_in[8];
    const float* bn0 = (const float*)d_in[9];
    const float* g1  = (const float*)d_in[10];
    const float* bn1 = (const float*)d_in[11];

    const int N = in_sizes[0] / 128;
    const int E = in_sizes[1] / 2;
    const int* src = ei;
    const int* dst = ei + E;

    float* dinv = (float*)d_ws;              // N
    float* coef = dinv + N;                  // E
    float* bufA = coef + E;                  // N*128
    float* bufB = bufA + (size_t)N * 128;    // N*128

    dim3 blk(256);
    int gN     = (N + 255) / 256;
    int gE     = (E + 255) / 256;
    int gRow   = (N + 127) / 128;
    int gWaveE = (E + 7) / 8;
    int gWaveN = (N + 7) / 8;
    int gNF128 = (N * 32 + 255) / 256;
    int gNF64  = (N * 16 + 255) / 256;

    // deg^{-1/2} and per-edge coefficients
    gcn_fill_deg<<<gN, blk, 0, stream>>>(dinv, N);
    gcn_deg_count<<<gE, blk, 0, stream>>>(dst, dinv, E);
    gcn_dinv<<<gN, blk, 0, stream>>>(dinv, N);
    gcn_coef<<<gE, blk, 0, stream>>>(src, dst, dinv, coef, E);

    // ---- layer 0: x @ W0 -> aggregate -> LN+ReLU ----
    gcn_gemm_wmma<128><<<gRow, blk, 0, stream>>>(x, W0, bufA, N);
    gcn_init_agg<128><<<gNF128, blk, 0, stream>>>(bufA, b0, dinv, bufB, N);
    gcn_scatter<128><<<gWaveE, blk, 0, stream>>>(src, dst, coef, bufA, bufB, E);
    gcn_norm_relu<<<gWaveN, blk, 0, stream>>>(bufB, g0, bn0, bufA, N);

    // ---- layer 1 ----
    gcn_gemm_wmma<128><<<gRow, blk, 0, stream>>>(bufA, W1, bufB, N);
    gcn_init_agg<128><<<gNF128, blk, 0, stream>>>(bufB, b1, dinv, bufA, N);
    gcn_scatter<128><<<gWaveE, blk, 0, stream>>>(src, dst, coef, bufB, bufA, E);
    gcn_norm_relu<<<gWaveN, blk, 0, stream>>>(bufA, g1, bn1, bufB, N);

    // ---- layer 2 (128 -> 64), aggregate directly into d_out ----
    float* outf = (float*)d_out;
    gcn_gemm_wmma<64><<<gRow, blk, 0, stream>>>(bufB, W2, bufA, N);
    gcn_init_agg<64><<<gNF64, blk, 0, stream>>>(bufA, b2, dinv, outf, N);
    gcn_scatter<64><<<gWaveE, blk, 0, stream>>>(src, dst, coef, bufA, outf, E);
}